// GCN_57612691308664
// MI455X (gfx1250) — compile-verified
//
#include <hip/hip_runtime.h>
#include <hip/hip_bf16.h>
#include <math.h>

// ---------------------------------------------------------------------------
// GCN on MI455X (gfx1250).  Pipeline:
//   deg -> dinv                          (atomic float add, rsqrt)
//   h1  = x @ W1^T                       (WMMA f32 16x16x4)
//   agg1 = scatter(norm * gather(h1))    (L2-resident atomics, 16 lanes/edge)
//   a1  = relu(agg1 + h1*dinv^2 + b1)
//   h2  = a1 @ W2^T                      (WMMA f32 16x16x4)
//   agg2, a2 = relu(...)
//   out = softmax(a2 @ Wl^T + bl)
// ---------------------------------------------------------------------------

typedef __attribute__((ext_vector_type(2))) float v2f;
typedef __attribute__((ext_vector_type(8))) float v8f;

#define F_IN 37
#define HDIM 64

// ---------------- degree / dinv ----------------
__global__ void deg_kernel(const int* __restrict__ dst, float* __restrict__ deg, int E) {
    int e = blockIdx.x * blockDim.x + threadIdx.x;
    if (e < E) atomicAdd(&deg[dst[e]], 1.0f);
}

__global__ void dinv_kernel(float* __restrict__ d, int N) {
    int i = blockIdx.x * blockDim.x + threadIdx.x;
    if (i < N) d[i] = rsqrtf(d[i] + 1.0f);   // deg includes self-loop
}

// ---------------- dense transform: H[n,j] = sum_k X[n,k] * W[j,k] ----------
// One block = 256 threads = 8 waves; block covers 128 rows, wave covers 16
// rows x all 64 output columns via 4 accumulator tiles of V_WMMA_F32_16X16X4.
template <int K, int LDK>
__global__ __launch_bounds__(256)
void gemm_xWT(const float* __restrict__ X, const float* __restrict__ W,
              float* __restrict__ H, int N) {
    __shared__ float sX[128 * LDK];
    __shared__ float sW[64 * LDK];

    const int tid  = threadIdx.x;
    const int row0 = blockIdx.x * 128;

    // Stage W [64,K] into LDS (padded to LDK with zeros).
    for (int i = tid; i < 64 * K; i += 256) {
        int r = i / K, c = i - r * K;
        sW[r * LDK + c] = W[i];
    }
    if constexpr (LDK > K) {
        constexpr int P = LDK - K;
        for (int i = tid; i < 64 * P; i += 256) {
            int r = i / P, c = K + (i - r * P);
            sW[r * LDK + c] = 0.0f;
        }
    }
    // Stage X tile [128,K] into LDS (zero rows past N, zero K-padding).
    for (int i = tid; i < 128 * K; i += 256) {
        int r = i / K, c = i - r * K;
        int g = row0 + r;
        sX[r * LDK + c] = (g < N) ? X[(long)g * K + c] : 0.0f;
    }
    if constexpr (LDK > K) {
        constexpr int P = LDK - K;
        for (int i = tid; i < 128 * P; i += 256) {
            int r = i / P, c = K + (i - r * P);
            sX[r * LDK + c] = 0.0f;
        }
    }
    __syncthreads();

    const int wave = tid >> 5;          // 0..7 -> 16-row tile within block
    const int lane = tid & 31;
    const int half = lane >> 4;         // 0: K pair {0,1}, 1: K pair {2,3}
    const int l16  = lane & 15;
    const int m0   = wave * 16;

    v8f acc[4] = {};                    // output column tiles j = 0,16,32,48

    for (int k0 = 0; k0 < LDK; k0 += 4) {
        // A fragment (16x4): lane l16 = row M, half selects K pair.
        v2f a;
        a.x = sX[(m0 + l16) * LDK + k0 + 2 * half];
        a.y = sX[(m0 + l16) * LDK + k0 + 2 * half + 1];
#pragma unroll
        for (int jt = 0; jt < 4; ++jt) {
            // B fragment (4x16): B[k][j] = W[j][k]; lane l16 = column N.
            v2f b;
            b.x = sW[(jt * 16 + l16) * LDK + k0 + 2 * half];
            b.y = sW[(jt * 16 + l16) * LDK + k0 + 2 * half + 1];
            acc[jt] = __builtin_amdgcn_wmma_f32_16x16x4_f32(
                false, a, false, b, (short)0, acc[jt], false, false);
        }
    }

    // C/D layout: VGPR r -> M = r + 8*half, N = l16.
#pragma unroll
    for (int jt = 0; jt < 4; ++jt) {
#pragma unroll
        for (int r = 0; r < 8; ++r) {
            int row = row0 + m0 + half * 8 + r;
            if (row < N) H[(long)row * HDIM + jt * 16 + l16] = acc[jt][r];
        }
    }
}

// ---------------- edge aggregation -----------------------------------------
// 16 lanes cooperate on one edge: the 256B row gather is fully coalesced,
// per-edge weight recomputed from L2-resident dinv.  agg buffer (25.6MB) and
// h (25.6MB) both fit in the 192MB L2, so atomics resolve on-chip.
__global__ void agg_kernel(const float* __restrict__ H, const int* __restrict__ src,
                           const int* __restrict__ dst, const float* __restrict__ dinv,
                           float* agg, long total) {
    long gid = (long)blockIdx.x * blockDim.x + threadIdx.x;
    if (gid >= total) return;
    int e = (int)(gid >> 4);
    int c = (int)(gid & 15) << 2;
    int s = src[e], d = dst[e];
    float w = dinv[s] * dinv[d];
    const float4 hv = *(const float4*)(H + (long)s * HDIM + c);
    float* ap = agg + (long)d * HDIM + c;
    atomicAdd(ap + 0, hv.x * w);
    atomicAdd(ap + 1, hv.y * w);
    atomicAdd(ap + 2, hv.z * w);
    atomicAdd(ap + 3, hv.w * w);
}

// ---------------- self-loop + bias + relu ----------------------------------
__global__ void combine_relu(const float4* agg, const float4* h,
                             const float* __restrict__ dinv,
                             const float* __restrict__ bias,
                             float4* out, int N) {
    int i = blockIdx.x * blockDim.x + threadIdx.x;   // N*16 float4 groups
    if (i >= N * 16) return;
    int row = i >> 4;
    int c4  = (i & 15) << 2;
    float dn = dinv[row];
    float sn = dn * dn;
    float4 a = agg[i], hv = h[i];
    float4 o;
    o.x = fmaxf(a.x + hv.x * sn + bias[c4 + 0], 0.0f);
    o.y = fmaxf(a.y + hv.y * sn + bias[c4 + 1], 0.0f);
    o.z = fmaxf(a.z + hv.z * sn + bias[c4 + 2], 0.0f);
    o.w = fmaxf(a.w + hv.w * sn + bias[c4 + 3], 0.0f);
    out[i] = o;
}

// ---------------- final 64->3 projection + softmax -------------------------
__global__ void logits_softmax(const float* __restrict__ H, const float* __restrict__ Wl,
                               const float* __restrict__ bl, float* __restrict__ out, int N) {
    int n = blockIdx.x * blockDim.x + threadIdx.x;
    if (n >= N) return;
    float s0 = bl[0], s1 = bl[1], s2 = bl[2];
    const float4* hr = (const float4*)(H + (long)n * HDIM);
    const float4* w0 = (const float4*)(Wl);
    const float4* w1 = (const float4*)(Wl + HDIM);
    const float4* w2 = (const float4*)(Wl + 2 * HDIM);
#pragma unroll
    for (int i = 0; i < 16; ++i) {
        float4 h4 = hr[i], a = w0[i], b = w1[i], c = w2[i];
        s0 += h4.x * a.x + h4.y * a.y + h4.z * a.z + h4.w * a.w;
        s1 += h4.x * b.x + h4.y * b.y + h4.z * b.z + h4.w * b.w;
        s2 += h4.x * c.x + h4.y * c.y + h4.z * c.z + h4.w * c.w;
    }
    float m  = fmaxf(s0, fmaxf(s1, s2));
    float e0 = expf(s0 - m), e1 = expf(s1 - m), e2 = expf(s2 - m);
    float inv = 1.0f / (e0 + e1 + e2);
    out[n * 3 + 0] = e0 * inv;
    out[n * 3 + 1] = e1 * inv;
    out[n * 3 + 2] = e2 * inv;
}

extern "C" void kernel_launch(void* const* d_in, const int* in_sizes, int n_in,
                              void* d_out, int out_size, void* d_ws, size_t ws_size,
                              hipStream_t stream) {
    const float* x  = (const float*)d_in[0];
    const int*   ei = (const int*)d_in[1];      // [2, E] int32
    const float* W1 = (const float*)d_in[2];
    const float* b1 = (const float*)d_in[3];
    const float* W2 = (const float*)d_in[4];
    const float* b2 = (const float*)d_in[5];
    const float* Wl = (const float*)d_in[6];
    const float* bl = (const float*)d_in[7];

    const int N = in_sizes[0] / F_IN;
    const int E = in_sizes[1] / 2;
    const int* src = ei;
    const int* dst = ei + E;

    // Workspace layout (floats): dinv [N], then two N*64 ping-pong buffers.
    const long NH    = (long)N * HDIM;
    const long Npad  = ((long)N + 255) & ~255L;
    float* dinv = (float*)d_ws;
    float* bufA = dinv + Npad;
    float* bufB = bufA + NH;

    const int T = 256;
    const int gN    = (N + T - 1) / T;
    const int gE    = (E + T - 1) / T;
    const int gRows = (N + 127) / 128;
    const long aggT = (long)E * 16;
    const int gAgg  = (int)((aggT + T - 1) / T);
    const int gVec  = (N * 16 + T - 1) / T;

    // deg -> dinv
    hipMemsetAsync(dinv, 0, (size_t)N * sizeof(float), stream);
    deg_kernel<<<gE, T, 0, stream>>>(dst, dinv, E);
    dinv_kernel<<<gN, T, 0, stream>>>(dinv, N);

    // ---- layer 1 ----
    gemm_xWT<F_IN, 40><<<gRows, T, 0, stream>>>(x, W1, bufA, N);          // h1 -> A
    hipMemsetAsync(bufB, 0, (size_t)NH * sizeof(float), stream);
    agg_kernel<<<gAgg, T, 0, stream>>>(bufA, src, dst, dinv, bufB, aggT); // agg1 -> B
    combine_relu<<<gVec, T, 0, stream>>>((const float4*)bufB, (const float4*)bufA,
                                         dinv, b1, (float4*)bufA, N);     // a1 -> A

    // ---- layer 2 ----
    gemm_xWT<HDIM, HDIM><<<gRows, T, 0, stream>>>(bufA, W2, bufB, N);     // h2 -> B
    hipMemsetAsync(bufA, 0, (size_t)NH * sizeof(float), stream);
    agg_kernel<<<gAgg, T, 0, stream>>>(bufB, src, dst, dinv, bufA, aggT); // agg2 -> A
    combine_relu<<<gVec, T, 0, stream>>>((const float4*)bufA, (const float4*)bufB,
                                         dinv, b2, (float4*)bufB, N);     // a2 -> B

    // ---- head ----
    logits_softmax<<<gN, T, 0, stream>>>(bufB, Wl, bl, (float*)d_out, N);
}